// GATv2Net_4312147165751
// MI455X (gfx1250) — compile-verified
//
#include <hip/hip_runtime.h>
#include <cmath>
#include <cstddef>
#include <cstdint>

typedef __attribute__((ext_vector_type(2))) float v2f;
typedef __attribute__((ext_vector_type(8))) float v8f;

#define F_IN   116
#define HID    32
#define HEADS  4
#define NCLS   2
#define NGRAPH 100
#define IN1    (F_IN + HID)   // 148
#define OC1    (HEADS * HID)  // 128
#define IN2    OC1            // 128
#define OC2    HID            // 32

#if __has_builtin(__builtin_amdgcn_global_load_async_to_lds_b32)
#define HAVE_ASYNC_LDS 1
typedef __attribute__((address_space(1))) int* gptr_i32;
typedef __attribute__((address_space(3))) int* lptr_i32;
#endif

// ---------- atomics ----------
__device__ __forceinline__ void atomicAddF(float* p, float v) {
  unsafeAtomicAdd(p, v);  // global_atomic_add_f32
}

// float atomic max via monotonic int bit-trick (init must be -inf)
__device__ __forceinline__ void atomicMaxF(float* addr, float val) {
  if (!(val == val)) return;
  int iv = __float_as_int(val);
  if (iv >= 0) atomicMax((int*)addr, iv);
  else atomicMin((unsigned int*)addr, (unsigned int)iv);
}

__device__ __forceinline__ void wait_async0() {
#ifdef HAVE_ASYNC_LDS
#if __has_builtin(__builtin_amdgcn_s_wait_asynccnt)
  __builtin_amdgcn_s_wait_asynccnt(0);
#else
  asm volatile("s_wait_asynccnt 0x0" ::: "memory");
#endif
#endif
}

// ---------- utility kernels ----------
__global__ void fill_kernel(float* __restrict__ p, float v, size_t n) {
  size_t t = (size_t)blockIdx.x * blockDim.x + threadIdx.x;
  if (t < n) p[t] = v;
}

// h[:,0:116] = x ; h[:,116:148] = 0
__global__ void pack_x(const float* __restrict__ x, float* __restrict__ h, int N) {
  size_t t = (size_t)blockIdx.x * blockDim.x + threadIdx.x;
  if (t >= (size_t)N * IN1) return;
  int c = (int)(t % IN1);
  size_t n = t / IN1;
  h[t] = (c < F_IN) ? x[n * F_IN + c] : 0.0f;
}

// edge_msg scatter: h[dst, 116+c] += edge_attr[e]*We[c] + be[c]
__global__ void edge_msg_kernel(const float* __restrict__ ea, const int* __restrict__ dst,
                                const float* __restrict__ We, const float* __restrict__ be,
                                float* __restrict__ h, int E) {
  size_t t = (size_t)blockIdx.x * blockDim.x + threadIdx.x;
  if (t >= (size_t)E * HID) return;
  int e = (int)(t >> 5);
  int c = (int)(t & 31);
  float v = ea[e] * We[c] + be[c];
  atomicAddF(&h[(size_t)dst[e] * IN1 + F_IN + c], v);
}

// ---------- FP32 WMMA GEMM: C[N,OC] = A[N,K] @ W[K,OC] + bias ----------
// One block per 16-row strip; blockDim/32 waves each own a 16-col tile.
// A strip staged once into LDS (async global->LDS when available), odd row
// stride for conflict-free ds_load. K stepped by 4 (V_WMMA_F32_16X16X4_F32):
// A frag: lanes 0-15 hold {K=k0,k0+1}, lanes 16-31 {k0+2,k0+3}; B mirrored
// over N; C/D: VGPR r -> row M=r (lanes 0-15) / M=8+r (lanes 16-31).
__global__ void gemm_wmma_f32(const float* __restrict__ A, const float* __restrict__ W,
                              const float* __restrict__ bias, float* __restrict__ C,
                              int N, int K, int OC) {
  extern __shared__ float sA[];        // 16 x K, row stride (K|1)
  const int LDA = K | 1;
  const int tileM = blockIdx.x * 16;
  const int wid  = (int)(threadIdx.x >> 5);
  const int lane = (int)(threadIdx.x & 31);
  const int tileN = wid * 16;

  // ---- cooperative stage of 16xK A strip into LDS ----
  const int elems = 16 * K;
  for (int i = (int)threadIdx.x; i < elems; i += (int)blockDim.x) {
    int r = i / K, k = i - r * K;
    int row = tileM + r;
    if (row >= N) row = N - 1;          // clamp (stores are guarded later)
#ifdef HAVE_ASYNC_LDS
    __builtin_amdgcn_global_load_async_to_lds_b32(
        (gptr_i32)(A + (size_t)row * K + k),
        (lptr_i32)(sA + r * LDA + k), 0, 0);
#else
    sA[r * LDA + k] = A[(size_t)row * K + k];
#endif
  }
  wait_async0();
  __syncthreads();

  if (tileN >= OC) return;              // uniform per wave
  const int half = lane >> 4;           // 0 or 1
  const int l15  = lane & 15;
  const float* arow = sA + l15 * LDA;               // LDS
  const float* wcol = W + (size_t)(tileN + l15);    // global (L2/WGP$ resident)
  v8f acc = {};
  for (int k0 = 0; k0 < K; k0 += 4) {
    const int ka = k0 + 2 * half;
    v2f a, b;
    a.x = arow[ka];
    a.y = arow[ka + 1];
    b.x = wcol[(size_t)ka * OC];
    b.y = wcol[(size_t)(ka + 1) * OC];
    acc = __builtin_amdgcn_wmma_f32_16x16x4_f32(false, a, false, b, (short)0, acc,
                                                false, false);
  }
  const int col = tileN + l15;
  const float bv = bias[col];
  const int rowBase = tileM + half * 8;
#pragma unroll
  for (int r = 0; r < 8; ++r) {
    int row = rowBase + r;
    if (row < N) C[(size_t)row * OC + col] = acc[r] + bv;
  }
}

// ---------- GATv2 attention ----------
// One wave32 per edge, lane = channel. Per-lane partials for all H heads,
// then a 5-step shfl_xor butterfly reduces across the 32 channels.
template <int H>
__global__ __launch_bounds__(256)
void att_logits_wave(const int* __restrict__ src, const int* __restrict__ dst,
                     const float* __restrict__ xl, const float* __restrict__ xr,
                     const float* __restrict__ att, float* __restrict__ elog,
                     float* __restrict__ emax, int E, int T) {
  const int idx  = (int)((blockIdx.x * blockDim.x + threadIdx.x) >> 5);
  const int lane = (int)(threadIdx.x & 31);
  if (idx >= T) return;
  const int s = idx < E ? src[idx] : idx - E;   // self-loops appended after E
  const int d = idx < E ? dst[idx] : idx - E;
  const int HC = H * HID;
  float part[H];
#pragma unroll
  for (int hh = 0; hh < H; ++hh) {
    float v = xl[(size_t)s * HC + hh * HID + lane] +
              xr[(size_t)d * HC + hh * HID + lane];
    v = v > 0.0f ? v : 0.2f * v;
    part[hh] = att[hh * HID + lane] * v;
  }
#pragma unroll
  for (int off = 16; off > 0; off >>= 1) {
#pragma unroll
    for (int hh = 0; hh < H; ++hh) part[hh] += __shfl_xor(part[hh], off, 32);
  }
  if (lane == 0) {
#pragma unroll
    for (int hh = 0; hh < H; ++hh) {
      elog[(size_t)idx * H + hh] = part[hh];
      atomicMaxF(&emax[(size_t)d * H + hh], part[hh]);
    }
  }
}

// ez = exp(e - emax[dst]); denom[dst] += ez  (elog now holds unnormalized ez)
template <int H>
__global__ void att_exp(const int* __restrict__ dst, const float* __restrict__ emax,
                        float* __restrict__ elog, float* __restrict__ den,
                        int E, int T) {
  long long t = (long long)blockIdx.x * blockDim.x + threadIdx.x;
  if (t >= (long long)T * H) return;
  int idx = (int)(t / H);
  int hh  = (int)(t % H);
  int d = idx < E ? dst[idx] : idx - E;
  float ez = __expf(elog[t] - emax[(size_t)d * H + hh]);
  elog[t] = ez;
  atomicAddF(&den[(size_t)d * H + hh], ez);
}

// out[d, h*C+c] += ez * xl[s, h*C+c]   (normalization folded into next pass)
template <int H>
__global__ void att_agg(const int* __restrict__ src, const int* __restrict__ dst,
                        const float* __restrict__ elog,
                        const float* __restrict__ xl, float* __restrict__ out, int E) {
  const int idx = blockIdx.x;
  const int t = (int)threadIdx.x;   // 0 .. H*HID-1
  const int hh = t / HID;
  int s = idx < E ? src[idx] : idx - E;
  int d = idx < E ? dst[idx] : idx - E;
  const int HC = H * HID;
  float w = elog[(size_t)idx * H + hh];
  atomicAddF(&out[(size_t)d * HC + t], w * xl[(size_t)s * HC + t]);
}

// x[n,c] = elu(x[n,c] / (den[n, c/HID] + 1e-16) + bias[c])
template <int H>
__global__ void norm_bias_elu(float* __restrict__ x, const float* __restrict__ den,
                              const float* __restrict__ bias, int N) {
  const int HC = H * HID;
  size_t t = (size_t)blockIdx.x * blockDim.x + threadIdx.x;
  if (t >= (size_t)N * HC) return;
  int c = (int)(t % HC);
  size_t n = t / HC;
  float v = x[t] / (den[n * H + c / HID] + 1e-16f) + bias[c];
  x[t] = v > 0.0f ? v : (__expf(v) - 1.0f);
}

// ---------- pooling + head ----------
__global__ void pool_kernel(const float* __restrict__ out2, const int* __restrict__ batch,
                            float* __restrict__ pooled, float* __restrict__ cnt, int N) {
  size_t t = (size_t)blockIdx.x * blockDim.x + threadIdx.x;
  if (t >= (size_t)N * HID) return;
  int n = (int)(t >> 5);
  int c = (int)(t & 31);
  int g = batch[n];
  atomicAddF(&pooled[(size_t)g * HID + c], out2[t]);
  if (c == 0) atomicAddF(&cnt[g], 1.0f);
}

__global__ void head_kernel(const float* __restrict__ pooled, const float* __restrict__ cnt,
                            const float* __restrict__ Wfc, const float* __restrict__ bfc,
                            float* __restrict__ out) {
  int g = (int)threadIdx.x;
  if (g >= NGRAPH) return;
  float inv = 1.0f / fmaxf(cnt[g], 1.0f);
  float l0 = bfc[0], l1 = bfc[1];
#pragma unroll
  for (int c = 0; c < HID; ++c) {
    float p = pooled[(size_t)g * HID + c] * inv;
    l0 += p * Wfc[c * NCLS + 0];
    l1 += p * Wfc[c * NCLS + 1];
  }
  float m = fmaxf(l0, l1);
  float lse = m + __logf(__expf(l0 - m) + __expf(l1 - m));
  out[g * NCLS + 0] = l0 - lse;
  out[g * NCLS + 1] = l1 - lse;
}

// ---------- launcher ----------
static inline unsigned nblk(size_t total, int bs) {
  return (unsigned)((total + (size_t)bs - 1) / bs);
}

extern "C" void kernel_launch(void* const* d_in, const int* in_sizes, int n_in,
                              void* d_out, int out_size, void* d_ws, size_t ws_size,
                              hipStream_t stream) {
  (void)n_in; (void)out_size; (void)ws_size;
  const float* x         = (const float*)d_in[0];
  const float* edge_attr = (const float*)d_in[1];
  const int*   eidx      = (const int*)d_in[2];
  const int*   batch     = (const int*)d_in[3];
  const float* We   = (const float*)d_in[4];
  const float* be   = (const float*)d_in[5];
  const float* W1l  = (const float*)d_in[6];
  const float* b1l  = (const float*)d_in[7];
  const float* W1r  = (const float*)d_in[8];
  const float* b1r  = (const float*)d_in[9];
  const float* att1 = (const float*)d_in[10];
  const float* bias1= (const float*)d_in[11];
  const float* W2l  = (const float*)d_in[12];
  const float* b2l  = (const float*)d_in[13];
  const float* W2r  = (const float*)d_in[14];
  const float* b2r  = (const float*)d_in[15];
  const float* att2 = (const float*)d_in[16];
  const float* bias2= (const float*)d_in[17];
  const float* Wfc  = (const float*)d_in[18];
  const float* bfc  = (const float*)d_in[19];

  const int N = in_sizes[0] / F_IN;   // 50000
  const int E = in_sizes[2] / 2;      // 800000
  const int T = E + N;                // edges + self-loops
  const int* srcI = eidx;             // edge_index[0,:]
  const int* dstI = eidx + E;         // edge_index[1,:]

  // workspace layout (floats); conv2 buffers reuse the h region
  float* ws = (float*)d_ws;
  size_t off = 0;
  float* h    = ws + off; off += (size_t)N * IN1;
  float* xl1  = ws + off; off += (size_t)N * OC1;
  float* xr1  = ws + off; off += (size_t)N * OC1;
  float* out1 = ws + off; off += (size_t)N * OC1;
  float* elog = ws + off; off += (size_t)T * HEADS;
  float* emax = ws + off; off += (size_t)N * HEADS;
  float* den  = ws + off; off += (size_t)N * HEADS;
  float* pooled = ws + off; off += (size_t)NGRAPH * HID;
  float* cnt  = ws + off; off += NGRAPH;
  float* xl2  = h;
  float* xr2  = h + (size_t)N * OC2;
  float* out2 = h + (size_t)2 * N * OC2;

  // 1) h = [x | scatter(edge_attr @ We + be)]
  pack_x<<<nblk((size_t)N * IN1, 256), 256, 0, stream>>>(x, h, N);
  edge_msg_kernel<<<nblk((size_t)E * HID, 256), 256, 0, stream>>>(edge_attr, dstI, We, be, h, E);

  // 2) conv1 GEMMs (fp32 WMMA; 8 waves/block share an LDS-staged A strip)
  const unsigned gRows = (unsigned)((N + 15) / 16);
  const size_t shm1 = (size_t)16 * (IN1 | 1) * sizeof(float);
  gemm_wmma_f32<<<gRows, (OC1 / 16) * 32, shm1, stream>>>(h, W1l, b1l, xl1, N, IN1, OC1);
  gemm_wmma_f32<<<gRows, (OC1 / 16) * 32, shm1, stream>>>(h, W1r, b1r, xr1, N, IN1, OC1);

  // 3) conv1 attention (segment softmax over dst) + aggregate
  fill_kernel<<<nblk((size_t)N * HEADS, 256), 256, 0, stream>>>(emax, -INFINITY, (size_t)N * HEADS);
  fill_kernel<<<nblk((size_t)N * HEADS, 256), 256, 0, stream>>>(den, 0.0f, (size_t)N * HEADS);
  fill_kernel<<<nblk((size_t)N * OC1, 256), 256, 0, stream>>>(out1, 0.0f, (size_t)N * OC1);
  att_logits_wave<HEADS><<<nblk((size_t)T * 32, 256), 256, 0, stream>>>(srcI, dstI, xl1, xr1, att1, elog, emax, E, T);
  att_exp<HEADS><<<nblk((size_t)T * HEADS, 256), 256, 0, stream>>>(dstI, emax, elog, den, E, T);
  att_agg<HEADS><<<T, HEADS * HID, 0, stream>>>(srcI, dstI, elog, xl1, out1, E);
  norm_bias_elu<HEADS><<<nblk((size_t)N * OC1, 256), 256, 0, stream>>>(out1, den, bias1, N);

  // 4) conv2 GEMMs (fp32 WMMA), heads=1
  const size_t shm2 = (size_t)16 * (IN2 | 1) * sizeof(float);
  gemm_wmma_f32<<<gRows, (OC2 / 16) * 32, shm2, stream>>>(out1, W2l, b2l, xl2, N, IN2, OC2);
  gemm_wmma_f32<<<gRows, (OC2 / 16) * 32, shm2, stream>>>(out1, W2r, b2r, xr2, N, IN2, OC2);

  // 5) conv2 attention + aggregate
  fill_kernel<<<nblk((size_t)N, 256), 256, 0, stream>>>(emax, -INFINITY, (size_t)N);
  fill_kernel<<<nblk((size_t)N, 256), 256, 0, stream>>>(den, 0.0f, (size_t)N);
  fill_kernel<<<nblk((size_t)N * OC2, 256), 256, 0, stream>>>(out2, 0.0f, (size_t)N * OC2);
  att_logits_wave<1><<<nblk((size_t)T * 32, 256), 256, 0, stream>>>(srcI, dstI, xl2, xr2, att2, elog, emax, E, T);
  att_exp<1><<<nblk((size_t)T, 256), 256, 0, stream>>>(dstI, emax, elog, den, E, T);
  att_agg<1><<<T, HID, 0, stream>>>(srcI, dstI, elog, xl2, out2, E);
  norm_bias_elu<1><<<nblk((size_t)N * OC2, 256), 256, 0, stream>>>(out2, den, bias2, N);

  // 6) mean pool + classifier head + log_softmax
  fill_kernel<<<nblk((size_t)NGRAPH * HID + NGRAPH, 256), 256, 0, stream>>>(pooled, 0.0f, (size_t)NGRAPH * HID + NGRAPH);
  pool_kernel<<<nblk((size_t)N * HID, 256), 256, 0, stream>>>(out2, batch, pooled, cnt, N);
  head_kernel<<<1, 128, 0, stream>>>(pooled, cnt, Wfc, bfc, (float*)d_out);
}